// KeyValueAttention_49830210568541
// MI455X (gfx1250) — compile-verified
//
#include <hip/hip_runtime.h>

// Problem constants (from reference)
#define B_    32
#define T_    2048
#define F_    2048
#define DQ_   1024
#define UNITS 1024
#define HALF_ 1024   // F/2
#define CDIM  2048   // HALF_ + DQ_

typedef __attribute__((ext_vector_type(2))) float v2f;
typedef __attribute__((ext_vector_type(8))) float v8f;

__device__ __forceinline__ float waveReduceSum(float v) {
#pragma unroll
  for (int off = 16; off > 0; off >>= 1)
    v += __shfl_xor(v, off, 32);
  return v;
}

// ---------------------------------------------------------------------------
// Kernel 1: w[c] = sum_u W1[c,u] * V[u]   (c in [0, 2048))
// One wave32 per output element; coalesced row reads; shuffle reduce.
// ---------------------------------------------------------------------------
__global__ void kv_attn_compute_w(const float* __restrict__ W1,
                                  const float* __restrict__ V,
                                  float* __restrict__ wbuf) {
  const int c    = blockIdx.x * 8 + (threadIdx.x >> 5);
  const int lane = threadIdx.x & 31;
  const float* row = W1 + (size_t)c * UNITS;
  float acc = 0.0f;
#pragma unroll 8
  for (int i = lane; i < UNITS; i += 32)
    acc = fmaf(row[i], V[i], acc);
  acc = waveReduceSum(acc);
  if (lane == 0) wbuf[c] = acc;
}

// ---------------------------------------------------------------------------
// Kernel 2: qdotc[b] = query[b,:]·w[1024:2048] + b1·V + bV
// One 256-thread block per batch.
// ---------------------------------------------------------------------------
__global__ void kv_attn_qdot(const float* __restrict__ query,
                             const float* __restrict__ wbuf,
                             const float* __restrict__ b1,
                             const float* __restrict__ V,
                             const float* __restrict__ bV,
                             float* __restrict__ qdotc) {
  const int b   = blockIdx.x;
  const int tid = threadIdx.x;
  __shared__ float red[256];
  float acc = 0.0f;
  for (int i = tid; i < DQ_; i += 256)
    acc = fmaf(query[b * DQ_ + i], wbuf[HALF_ + i], acc);
  for (int i = tid; i < UNITS; i += 256)
    acc = fmaf(b1[i], V[i], acc);
  red[tid] = acc;
  __syncthreads();
  for (int o = 128; o > 0; o >>= 1) {
    if (tid < o) red[tid] += red[tid + o];
    __syncthreads();
  }
  if (tid == 0) qdotc[b] = red[0] + bV[0];
}

// ---------------------------------------------------------------------------
// Kernel 3: scores[b,t] = values[b,t,0:1024]·w[0:1024] + qdotc[b]
// One wave32 per (b,t) row; float4 coalesced loads; shuffle reduce.
// Grid: (B*T/8) blocks of 256 threads.
// ---------------------------------------------------------------------------
__global__ void kv_attn_scores(const float* __restrict__ values,
                               const float* __restrict__ wbuf,
                               const float* __restrict__ qdotc,
                               float* __restrict__ scores) {
  const int wave = blockIdx.x * 8 + (threadIdx.x >> 5);
  const int lane = threadIdx.x & 31;
  const int b = wave >> 11;          // T_ == 2048
  const int t = wave & (T_ - 1);
  const float4* row = (const float4*)(values + ((size_t)b * T_ + t) * F_);
  const float4* w4  = (const float4*)wbuf;
  float acc = 0.0f;
#pragma unroll
  for (int i = lane; i < HALF_ / 4; i += 32) {
    float4 x = row[i];
    float4 y = w4[i];
    acc = fmaf(x.x, y.x, fmaf(x.y, y.y, fmaf(x.z, y.z, fmaf(x.w, y.w, acc))));
  }
  acc = waveReduceSum(acc);
  if (lane == 0) scores[wave] = acc + qdotc[b];
}

// ---------------------------------------------------------------------------
// Kernel 4: softmax over T per batch; writes attention weights.
// One 256-thread block per batch, 8 elements per thread.
// ---------------------------------------------------------------------------
__global__ void kv_attn_softmax(const float* __restrict__ scores,
                                float* __restrict__ aw) {
  const int b   = blockIdx.x;
  const int tid = threadIdx.x;
  __shared__ float red[256];
  const float* s = scores + b * T_;
  float v[8];
  float m = -3.402823466e38f;
#pragma unroll
  for (int i = 0; i < 8; ++i) {
    v[i] = s[tid + 256 * i];
    m = fmaxf(m, v[i]);
  }
  red[tid] = m;
  __syncthreads();
  for (int o = 128; o > 0; o >>= 1) {
    if (tid < o) red[tid] = fmaxf(red[tid], red[tid + o]);
    __syncthreads();
  }
  m = red[0];
  __syncthreads();
  float sum = 0.0f;
#pragma unroll
  for (int i = 0; i < 8; ++i) {
    v[i] = __expf(v[i] - m);
    sum += v[i];
  }
  red[tid] = sum;
  __syncthreads();
  for (int o = 128; o > 0; o >>= 1) {
    if (tid < o) red[tid] += red[tid + o];
    __syncthreads();
  }
  const float inv = 1.0f / red[0];
#pragma unroll
  for (int i = 0; i < 8; ++i)
    aw[b * T_ + tid + 256 * i] = v[i] * inv;
}

// ---------------------------------------------------------------------------
// Kernel 5: context partials via V_WMMA_F32_16X16X4_F32.
//   out[(b*S+s)*F + f] = sum_{t in chunk s} aw[b,t] * values[b,t,f]
// One wave per (b, s, 64-wide f tile). Per k-step (4 t's) the wave issues
// two coalesced float4 B loads (256B contiguous per half-wave) and 4 WMMAs
// whose N dim covers f0+4n+j (j = accumulator index), M replicated via the
// broadcast A operand (aw pair per lane half).
// ---------------------------------------------------------------------------
__global__ void kv_attn_context_wmma(const float* __restrict__ aw,
                                     const float* __restrict__ values,
                                     float* __restrict__ out,
                                     int S, int Tchunk) {
  const int wave = blockIdx.x * 8 + (threadIdx.x >> 5);
  const int lane = threadIdx.x & 31;
  const int ftile = wave & 31;            // 32 tiles of 64 floats = F_
  const int s     = (wave >> 5) % S;
  const int b     = wave / (32 * S);
  const int f0    = ftile * 64;
  const int g     = lane >> 4;            // half-wave: K offset 0 or 2
  const int n     = lane & 15;            // N index within tile
  const int t0 = s * Tchunk;
  const int t1 = t0 + Tchunk;

  v8f acc0 = {0,0,0,0,0,0,0,0};
  v8f acc1 = {0,0,0,0,0,0,0,0};
  v8f acc2 = {0,0,0,0,0,0,0,0};
  v8f acc3 = {0,0,0,0,0,0,0,0};

  const float* awb = aw + b * T_;
  const float* vb  = values + (size_t)b * T_ * F_ + f0 + 4 * n;

  for (int t = t0; t < t1; t += 4) {
    const int tr = t + 2 * g;             // this half-wave's K rows: tr, tr+1
    v2f a = *(const v2f*)(awb + tr);      // A: {aw[tr], aw[tr+1]} (M-replicated)
    const float* p = vb + (size_t)tr * F_;
    float4 r0 = *(const float4*)p;        // B row tr,   f = f0+4n .. f0+4n+3
    float4 r1 = *(const float4*)(p + F_); // B row tr+1, same f's
    v2f b0 = {r0.x, r1.x};
    v2f b1v = {r0.y, r1.y};
    v2f b2 = {r0.z, r1.z};
    v2f b3 = {r0.w, r1.w};
    acc0 = __builtin_amdgcn_wmma_f32_16x16x4_f32(false, a, false, b0,  (short)0, acc0, false, false);
    acc1 = __builtin_amdgcn_wmma_f32_16x16x4_f32(false, a, false, b1v, (short)0, acc1, false, false);
    acc2 = __builtin_amdgcn_wmma_f32_16x16x4_f32(false, a, false, b2,  (short)0, acc2, false, false);
    acc3 = __builtin_amdgcn_wmma_f32_16x16x4_f32(false, a, false, b3,  (short)0, acc3, false, false);
  }

  // D[m,n]: VGPR v holds M=v+8g rows; all M rows identical, take VGPR0 of
  // lanes 0..15 -> ctx[f0+4n+j] for accumulator j. Pack into one float4 store.
  if (lane < 16) {
    float4 o;
    o.x = acc0[0]; o.y = acc1[0]; o.z = acc2[0]; o.w = acc3[0];
    *(float4*)(out + (size_t)(b * S + s) * F_ + f0 + 4 * n) = o;
  }
}

// ---------------------------------------------------------------------------
// Kernel 6: reduce T-split partials into the context output.
// ---------------------------------------------------------------------------
__global__ void kv_attn_reduce_ctx(const float* __restrict__ partial,
                                   float* __restrict__ ctx, int S) {
  const int i = blockIdx.x * 256 + threadIdx.x;   // [0, B_*F_)
  const int b = i >> 11;                          // F_ == 2048
  const int f = i & (F_ - 1);
  float s = 0.0f;
  for (int k = 0; k < S; ++k)
    s += partial[(size_t)(b * S + k) * F_ + f];
  ctx[i] = s;
}

// ---------------------------------------------------------------------------
extern "C" void kernel_launch(void* const* d_in, const int* in_sizes, int n_in,
                              void* d_out, int out_size, void* d_ws, size_t ws_size,
                              hipStream_t stream) {
  (void)in_sizes; (void)n_in; (void)out_size;
  const float* query  = (const float*)d_in[0];   // (32, 1024)
  const float* values = (const float*)d_in[1];   // (32, 2048, 2048)
  const float* W1     = (const float*)d_in[2];   // (2048, 1024)
  const float* b1     = (const float*)d_in[3];   // (1024,)
  const float* V      = (const float*)d_in[4];   // (1024, 1)
  const float* bV     = (const float*)d_in[5];   // (1,)

  float* ctx = (float*)d_out;                    // (32, 2048) context
  float* aw  = (float*)d_out + (size_t)B_ * F_;  // (32, 2048) attention weights

  float* ws     = (float*)d_ws;
  float* wbuf   = ws;            // 2048 floats: W1 @ V
  float* qdotc  = ws + 2048;     // 32 floats
  float* partial = ws + 4096;    // B_*S*F_ floats (T-split context partials)
  // Scores use the ctx half of d_out as scratch (consumed by softmax before
  // the context kernels overwrite it). Keeps mandatory ws usage tiny.
  float* scores = ctx;

  // Deterministic T-split factor based on available workspace.
  int S = 8;
  while (S > 1 &&
         (size_t)4096 * 4 + (size_t)B_ * S * F_ * 4 > ws_size)
    S >>= 1;
  const bool have_partial =
      ((size_t)4096 * 4 + (size_t)B_ * S * F_ * 4) <= ws_size && S > 1;
  if (!have_partial) S = 1;

  kv_attn_compute_w<<<CDIM / 8, 256, 0, stream>>>(W1, V, wbuf);
  kv_attn_qdot<<<B_, 256, 0, stream>>>(query, wbuf, b1, V, bV, qdotc);
  kv_attn_scores<<<(B_ * T_) / 8, 256, 0, stream>>>(values, wbuf, qdotc, scores);
  kv_attn_softmax<<<B_, 256, 0, stream>>>(scores, aw);

  if (S > 1) {
    kv_attn_context_wmma<<<128 * S, 256, 0, stream>>>(aw, values, partial, S, T_ / S);
    kv_attn_reduce_ctx<<<(B_ * F_) / 256, 256, 0, stream>>>(partial, ctx, S);
  } else {
    // Writes directly into ctx (same layout with S == 1).
    kv_attn_context_wmma<<<128, 256, 0, stream>>>(aw, values, ctx, 1, T_);
  }
}